// GCN_POOL_46394236731692
// MI455X (gfx1250) — compile-verified
//
#include <hip/hip_runtime.h>
#include <math.h>

typedef float v2f __attribute__((ext_vector_type(2)));
typedef float v8f __attribute__((ext_vector_type(8)));

#define DIM 128   // length == dim == 128 in this problem

// ---------------- degree / normalization ----------------

__global__ void k_deg_init(unsigned* __restrict__ deg, int N) {
    int i = blockIdx.x * 256 + threadIdx.x;
    if (i < N) deg[i] = 1u;                       // self-loop
}

__global__ void k_deg_count(const int* __restrict__ dst, unsigned* __restrict__ deg, int E) {
    int e = blockIdx.x * 256 + threadIdx.x;
    if (e < E) atomicAdd(&deg[dst[e]], 1u);
}

__global__ void k_dinv(float* __restrict__ dinv, int N) {   // in-place u32 count -> f32 rsqrt
    int i = blockIdx.x * 256 + threadIdx.x;
    if (i < N) {
        unsigned d = ((const unsigned*)dinv)[i];
        dinv[i] = rsqrtf((float)d);
    }
}

// ---------------- xw = x @ W via V_WMMA_F32_16X16X4_F32 ----------------
// One wave per 16x16 output tile. Also initializes out with the self-loop
// contribution: out[i][:] = xw[i][:] * dinv[i]^2.

__global__ void k_gemm_wmma(const float* __restrict__ x, const float* __restrict__ W,
                            const float* __restrict__ dinv,
                            float* __restrict__ xw, float* __restrict__ out,
                            int N, int Mtiles) {
    const int wave = threadIdx.x >> 5;
    const int lane = threadIdx.x & 31;
    const int tile = blockIdx.x * 8 + wave;      // 8 waves per 256-thread block
    if (tile >= Mtiles * 8) return;              // wave-uniform: EXEC stays all-ones
    const int tm = tile >> 3;                    // tile row
    const int tn = tile & 7;                     // tile col (DIM/16 == 8)
    const int row0 = tm * 16, col0 = tn * 16;

    const int mr    = lane & 15;                 // A row / B col within tile
    const int khalf = (lane >> 4) << 1;          // 0 for lanes 0-15, 2 for lanes 16-31

    const int arow_i = min(row0 + mr, N - 1);    // clamp only matters for ragged last tile
    const float* __restrict__ arow = x + (size_t)arow_i * DIM;
    const float* __restrict__ bcol = W + (col0 + mr);

    v8f c = {0.f, 0.f, 0.f, 0.f, 0.f, 0.f, 0.f, 0.f};
    #pragma unroll 8
    for (int kk = 0; kk < DIM; kk += 4) {
        // A 16x4 f32 fragment: lane (m=mr) holds K = kk+khalf, kk+khalf+1  (8B aligned)
        v2f a = *(const v2f*)(arow + kk + khalf);
        // B 4x16 f32 fragment: lane (n=mr) holds same K pair, column-strided in W
        v2f b;
        b.x = bcol[(size_t)(kk + khalf) * DIM];
        b.y = bcol[(size_t)(kk + khalf + 1) * DIM];
        c = __builtin_amdgcn_wmma_f32_16x16x4_f32(false, a, false, b,
                                                  (short)0, c, false, false);
    }

    // C/D layout: VGPR r -> m = r + 8*(lane>=16), n = lane&15
    const int mbase = (lane >> 4) << 3;
    const int gcol  = col0 + (lane & 15);
    if (row0 + 16 <= N) {
        // full tile: wave-uniform fast path, no per-store EXEC games
        #pragma unroll
        for (int r = 0; r < 8; ++r) {
            const int grow = row0 + mbase + r;
            const float v  = c[r];
            const float di = dinv[grow];
            xw [(size_t)grow * DIM + gcol] = v;
            out[(size_t)grow * DIM + gcol] = v * di * di;   // self-loop init
        }
    } else {
        // ragged last tile (unused for N % 16 == 0, kept for generality)
        #pragma unroll
        for (int r = 0; r < 8; ++r) {
            const int grow = row0 + mbase + r;
            if (grow < N) {
                const float v  = c[r];
                const float di = dinv[grow];
                xw [(size_t)grow * DIM + gcol] = v;
                out[(size_t)grow * DIM + gcol] = v * di * di;
            }
        }
    }
}

// ---------------- message gather + scatter-add (one wave per edge) ----------------
// xw (25.6 MB) and out (25.6 MB) both fit in the 192 MB L2: gathers and the
// fp32 atomics resolve in L2, HBM only sees compulsory traffic.

__global__ void k_scatter(const int* __restrict__ src, const int* __restrict__ dst,
                          const float* __restrict__ dinv, const float* __restrict__ xw,
                          float* __restrict__ out, int E) {
    const int wave = threadIdx.x >> 5;
    const int lane = threadIdx.x & 31;
    const int e = blockIdx.x * 8 + wave;
    if (e >= E) return;
    const int s = src[e], t = dst[e];
    const float norm = dinv[s] * dinv[t];
    const float4 v = ((const float4*)(xw + (size_t)s * DIM))[lane];   // 512B/wave gather
    float* __restrict__ o = out + (size_t)t * DIM + (lane << 2);
    __hip_atomic_fetch_add(o + 0, v.x * norm, __ATOMIC_RELAXED, __HIP_MEMORY_SCOPE_AGENT);
    __hip_atomic_fetch_add(o + 1, v.y * norm, __ATOMIC_RELAXED, __HIP_MEMORY_SCOPE_AGENT);
    __hip_atomic_fetch_add(o + 2, v.z * norm, __ATOMIC_RELAXED, __HIP_MEMORY_SCOPE_AGENT);
    __hip_atomic_fetch_add(o + 3, v.w * norm, __ATOMIC_RELAXED, __HIP_MEMORY_SCOPE_AGENT);
}

// ---------------- bias + ReLU + softmax over feature axis (one wave per row) ----------------

__global__ void k_softmax(float* __restrict__ out, const float* __restrict__ bias, int N) {
    const int wave = threadIdx.x >> 5;
    const int lane = threadIdx.x & 31;
    const int row = blockIdx.x * 8 + wave;
    if (row >= N) return;
    float4* __restrict__ rowp = (float4*)(out + (size_t)row * DIM);
    float4 v  = rowp[lane];
    float4 bb = ((const float4*)bias)[lane];
    v.x = fmaxf(v.x + bb.x, 0.f);
    v.y = fmaxf(v.y + bb.y, 0.f);
    v.z = fmaxf(v.z + bb.z, 0.f);
    v.w = fmaxf(v.w + bb.w, 0.f);

    float m = fmaxf(fmaxf(v.x, v.y), fmaxf(v.z, v.w));
    #pragma unroll
    for (int off = 16; off > 0; off >>= 1) m = fmaxf(m, __shfl_xor(m, off, 32));

    v.x = expf(v.x - m); v.y = expf(v.y - m);
    v.z = expf(v.z - m); v.w = expf(v.w - m);
    float s = v.x + v.y + v.z + v.w;
    #pragma unroll
    for (int off = 16; off > 0; off >>= 1) s += __shfl_xor(s, off, 32);

    const float inv = 1.0f / s;
    v.x *= inv; v.y *= inv; v.z *= inv; v.w *= inv;
    rowp[lane] = v;
}

// ---------------- launch ----------------

extern "C" void kernel_launch(void* const* d_in, const int* in_sizes, int n_in,
                              void* d_out, int out_size, void* d_ws, size_t ws_size,
                              hipStream_t stream) {
    const float* x  = (const float*)d_in[0];
    const int*   ei = (const int*)  d_in[1];
    const float* W  = (const float*)d_in[2];
    const float* b  = (const float*)d_in[3];
    const int N = in_sizes[0] / DIM;
    const int E = in_sizes[1] / 2;
    const int* src = ei;        // edge_index[0]
    const int* dst = ei + E;    // edge_index[1]

    float* out  = (float*)d_out;
    float* dinv = (float*)d_ws;            // N floats (first used as u32 degree)
    float* xw   = dinv + N;                // N*DIM floats

    k_deg_init <<<(N + 255) / 256, 256, 0, stream>>>((unsigned*)dinv, N);
    k_deg_count<<<(E + 255) / 256, 256, 0, stream>>>(dst, (unsigned*)dinv, E);
    k_dinv     <<<(N + 255) / 256, 256, 0, stream>>>(dinv, N);

    const int Mtiles = (N + 15) / 16;
    k_gemm_wmma<<<(Mtiles * 8 + 7) / 8, 256, 0, stream>>>(x, W, dinv, xw, out, N, Mtiles);

    k_scatter  <<<(E + 7) / 8, 256, 0, stream>>>(src, dst, dinv, xw, out, E);
    k_softmax  <<<(N + 7) / 8, 256, 0, stream>>>(out, b, N);
}